// DGCNN_cls_5806795784772
// MI455X (gfx1250) — compile-verified
//
#include <hip/hip_runtime.h>

typedef __attribute__((ext_vector_type(2))) float v2f;
typedef __attribute__((ext_vector_type(8))) float v8f;

constexpr int   B_  = 8;
constexpr int   N_  = 2048;
constexpr int   K_  = 20;
constexpr int   M_  = 512;    // N/4
constexpr int   K2_ = 10;
constexpr int   E_  = 1024;
__device__ constexpr float BN_SCALE_ = 0.9999950000374998f;  // 1/sqrt(1+1e-5)
constexpr float NEG_INF_ = -3.0e38f;

__device__ __forceinline__ float lrelu_(float x) { return x >= 0.f ? x : 0.2f * x; }

__device__ __forceinline__ void atomicMaxFloat(float* addr, float val) {
    int* ia = (int*)addr;
    int old = __float_as_int(*addr);
    while (val > __int_as_float(old)) {
        int assumed = old;
        old = atomicCAS(ia, assumed, __float_as_int(val));
        if (old == assumed) break;
    }
}

__device__ __forceinline__ v8f wmma4(v2f a, v2f b, v8f c) {
    return __builtin_amdgcn_wmma_f32_16x16x4_f32(false, a, false, b,
                                                 (short)0, c, false, false);
}

// ---------------------------------------------------------------------------
// Generic fp32 WMMA GEMM: D(O x Nn) = A(O x Kd) @ X(Kd x Nn).
// One wave computes a 64x32 output tile: 4 M-subtiles x 2 N-subtiles
// (8 accumulators, 8 V_WMMA_F32_16X16X4_F32 per K-step of 4). The B fragments
// are shared across the 4 M-subtiles (4x less B traffic than a 16-row tile).
// Fast path: interior tile && Kd%4==0 -> unguarded software-pipelined loads.
// Guarded fallback for edge tiles (conv1 K=3, FC head Nn=8 / O=40).
// mode 0: store raw
// mode 1: store lrelu((d + bias[o]) * g[o]*s + bb[o])
// mode 2: no store; atomic-max of lrelu(d*g[o]*s + bb[o]) into vmax[b][o]
// mode 3: store d + bias[o]
// X element: X[b*xb + k*xsk + n*xsn];  D element: D[b*db + row*dsr + col*dsc]
// ---------------------------------------------------------------------------
__global__ void __launch_bounds__(32)
gemm16_wmma(const float* __restrict__ A, int O, int Kd,
            const float* __restrict__ X, long long xb, long long xsk, long long xsn, int Nn,
            float* __restrict__ D, long long db, long long dsr, long long dsc,
            const float* __restrict__ g, const float* __restrict__ bb,
            const float* __restrict__ bias, float* __restrict__ vmax, long long vb,
            int mode)
{
    const int lane  = threadIdx.x;        // 0..31
    const int half  = lane >> 4;          // 0/1
    const int l     = lane & 15;
    const int tileN = blockIdx.x * 32;
    const int tileM = blockIdx.y * 64;
    const int b     = blockIdx.z;
    const int kh    = half * 2;           // K sub-offset (0 or 2)

    const float* Xb = X + (long long)b * xb;

    v8f c[4][2];
#pragma unroll
    for (int ms = 0; ms < 4; ++ms) { c[ms][0] = (v8f){}; c[ms][1] = (v8f){}; }

    const bool full = (tileM + 64 <= O) && (tileN + 32 <= Nn) && ((Kd & 3) == 0);
    if (full) {
        const float* ap[4];
#pragma unroll
        for (int ms = 0; ms < 4; ++ms)
            ap[ms] = A + (long long)(tileM + ms * 16 + l) * Kd + kh;
        const float* bp0 = Xb + (long long)kh * xsk + (long long)(tileN + l) * xsn;
        const float* bp1 = bp0 + 16 * xsn;

        // software-pipelined: loads for step i+1 issue before WMMAs of step i
        v2f a[4], p, q;
#pragma unroll
        for (int ms = 0; ms < 4; ++ms) { a[ms].x = ap[ms][0]; a[ms].y = ap[ms][1]; }
        p.x = bp0[0];      p.y = bp0[xsk];
        q.x = bp1[0];      q.y = bp1[xsk];

        for (int kk = 4; kk < Kd; kk += 4) {
            v2f an[4], pn, qn;
#pragma unroll
            for (int ms = 0; ms < 4; ++ms) { an[ms].x = ap[ms][4]; an[ms].y = ap[ms][5]; }
            pn.x = bp0[4 * xsk];   pn.y = bp0[5 * xsk];
            qn.x = bp1[4 * xsk];   qn.y = bp1[5 * xsk];
#pragma unroll
            for (int ms = 0; ms < 4; ++ms) {
                c[ms][0] = wmma4(a[ms], p, c[ms][0]);
                c[ms][1] = wmma4(a[ms], q, c[ms][1]);
            }
#pragma unroll
            for (int ms = 0; ms < 4; ++ms) { a[ms] = an[ms]; ap[ms] += 4; }
            p = pn; q = qn;
            bp0 += 4 * xsk; bp1 += 4 * xsk;
        }
#pragma unroll
        for (int ms = 0; ms < 4; ++ms) {
            c[ms][0] = wmma4(a[ms], p, c[ms][0]);
            c[ms][1] = wmma4(a[ms], q, c[ms][1]);
        }
    } else {
        const int bn0 = tileN + l, bn1 = tileN + 16 + l;
        for (int kk = 0; kk < Kd; kk += 4) {
            const int k0 = kk + kh, k1 = k0 + 1;
            const bool k0ok = (k0 < Kd), k1ok = (k1 < Kd);
            v2f p, q;
            p.x = (bn0 < Nn && k0ok) ? Xb[(long long)k0 * xsk + (long long)bn0 * xsn] : 0.f;
            p.y = (bn0 < Nn && k1ok) ? Xb[(long long)k1 * xsk + (long long)bn0 * xsn] : 0.f;
            q.x = (bn1 < Nn && k0ok) ? Xb[(long long)k0 * xsk + (long long)bn1 * xsn] : 0.f;
            q.y = (bn1 < Nn && k1ok) ? Xb[(long long)k1 * xsk + (long long)bn1 * xsn] : 0.f;
#pragma unroll
            for (int ms = 0; ms < 4; ++ms) {
                const int am = tileM + ms * 16 + l;
                v2f a;
                a.x = (am < O && k0ok) ? A[(long long)am * Kd + k0] : 0.f;
                a.y = (am < O && k1ok) ? A[(long long)am * Kd + k1] : 0.f;
                c[ms][0] = wmma4(a, p, c[ms][0]);
                c[ms][1] = wmma4(a, q, c[ms][1]);
            }
        }
    }

    // ---- epilogue (4 M-subtiles x 2 N-subtiles) ----
#pragma unroll
    for (int ms = 0; ms < 4; ++ms) {
#pragma unroll
        for (int sub = 0; sub < 2; ++sub) {
            const int col = tileN + sub * 16 + l;
#pragma unroll
            for (int r = 0; r < 8; ++r) {
                const int row = tileM + ms * 16 + r + half * 8;  // uniform per half
                const float d = c[ms][sub][r];
                if (mode == 2) {
                    float y = NEG_INF_;
                    if (row < O && col < Nn) {
                        float t = d * (g[row] * BN_SCALE_) + bb[row];
                        y = lrelu_(t);
                    }
                    for (int m = 1; m < 16; m <<= 1)
                        y = fmaxf(y, __shfl_xor(y, m, 32));
                    if (l == 0 && row < O)
                        atomicMaxFloat(&vmax[(long long)b * vb + row], y);
                } else if (row < O && col < Nn) {
                    float y;
                    if (mode == 0) {
                        y = d;
                    } else if (mode == 3) {
                        y = d + (bias ? bias[row] : 0.f);
                    } else { // mode 1
                        float t = d + (bias ? bias[row] : 0.f);
                        y = lrelu_(t * (g[row] * BN_SCALE_) + bb[row]);
                    }
                    D[(long long)b * db + (long long)row * dsr + (long long)col * dsc] = y;
                }
            }
        }
    }
}

// Build A = [Wd ; Wc - Wd] (2O x C) from W (O x 2C)
__global__ void prep_edge_w(const float* __restrict__ W, float* __restrict__ A, int O, int C)
{
    int t = blockIdx.x * 256 + threadIdx.x;
    int total = 2 * O * C;
    if (t >= total) return;
    int row = t / C, cc = t % C;
    if (row < O) A[t] = W[(long long)row * 2 * C + cc];
    else {
        int o = row - O;
        A[t] = W[(long long)o * 2 * C + C + cc] - W[(long long)o * 2 * C + cc];
    }
}

// ---------------------------------------------------------------------------
// kNN: for each query q, top-k (largest neg-sq-dist, ties -> lowest index).
// X: db (C x Ndb per batch), Q: queries (C x Mq per batch)
// ---------------------------------------------------------------------------
__global__ void knn_kernel(const float* __restrict__ X, long long sxb,
                           const float* __restrict__ Q, long long sqb,
                           int C, int Ndb, int Mq, int k,
                           int* __restrict__ out, long long sob)
{
    __shared__ float xnorm[2048];
    const int b = blockIdx.y;
    const float* Xb = X + (long long)b * sxb;
    const float* Qb = Q + (long long)b * sqb;

    for (int n = threadIdx.x; n < Ndb; n += blockDim.x) {
        float s = 0.f;
        for (int cc = 0; cc < C; ++cc) { float v = Xb[(long long)cc * Ndb + n]; s += v * v; }
        xnorm[n] = s;
    }
    __syncthreads();

    const int q = blockIdx.x * blockDim.x + threadIdx.x;
    if (q >= Mq) return;

    float qn = 0.f;
    for (int cc = 0; cc < C; ++cc) { float v = Qb[(long long)cc * Mq + q]; qn += v * v; }

    float best[20]; int bidx[20];
    for (int j = 0; j < k; ++j) { best[j] = NEG_INF_; bidx[j] = 0; }

    for (int m = 0; m < Ndb; ++m) {
        float ip = 0.f;
        for (int cc = 0; cc < C; ++cc)
            ip += Qb[(long long)cc * Mq + q] * Xb[(long long)cc * Ndb + m];
        float nd = 2.f * ip - qn - xnorm[m];
        if (nd > best[k - 1]) {
            int p = k - 1;
            while (p > 0 && best[p - 1] < nd) {
                best[p] = best[p - 1]; bidx[p] = bidx[p - 1]; --p;
            }
            best[p] = nd; bidx[p] = m;
        }
    }
    int* ob = out + (long long)b * sob + (long long)q * k;
    for (int j = 0; j < k; ++j) ob[j] = bidx[j];
}

// edge_conv finish: out[b,o,n] = max_j lrelu(s_o*(t[o,idx]+u[o,n]) + bb_o)
__global__ void edge_gathermax(const float* __restrict__ tu, long long tub, int O, int Nn,
                               const int* __restrict__ idx, long long idxb, int k,
                               const float* __restrict__ g, const float* __restrict__ bb,
                               float* __restrict__ out, long long outb, long long osr)
{
    long long t = (long long)blockIdx.x * blockDim.x + threadIdx.x;
    const int b = blockIdx.y;
    if (t >= (long long)O * Nn) return;
    const int o = (int)(t / Nn), n = (int)(t % Nn);
    const float* tb = tu + (long long)b * tub;
    const float u   = tb[(long long)(O + o) * Nn + n];
    const float s   = g[o] * BN_SCALE_;
    const float bo  = bb[o];
    const int* ib   = idx + (long long)b * idxb + (long long)n * k;
    float mx = NEG_INF_;
    for (int j = 0; j < k; ++j) {
        float h = tb[(long long)o * Nn + ib[j]] + u;
        mx = fmaxf(mx, lrelu_(h * s + bo));
    }
    out[(long long)b * outb + (long long)o * osr + n] = mx;
}

// Farthest point sampling: one block (256 threads) per batch
__global__ void fps_kernel(const float* __restrict__ xyz, int* __restrict__ outIdx)
{
    __shared__ float dists[2048];
    __shared__ float redv[256];
    __shared__ int   redi[256];
    __shared__ int   s_last;
    const int b = blockIdx.x;
    const int tid = threadIdx.x;
    const float* px = xyz + (long long)b * 3 * N_;

    for (int n = tid; n < N_; n += 256) dists[n] = 1.0e10f;
    if (tid == 0) s_last = 0;
    __syncthreads();

    for (int i = 0; i < M_; ++i) {
        const int last = s_last;
        if (tid == 0) outIdx[b * M_ + i] = last;
        const float lx = px[last], ly = px[N_ + last], lz = px[2 * N_ + last];
        float bv = NEG_INF_; int bi = 0x7fffffff;
        for (int n = tid; n < N_; n += 256) {
            float dx = px[n] - lx, dy = px[N_ + n] - ly, dz = px[2 * N_ + n] - lz;
            float nd = fminf(dists[n], dx * dx + dy * dy + dz * dz);
            dists[n] = nd;
            if (nd > bv) { bv = nd; bi = n; }
        }
        redv[tid] = bv; redi[tid] = bi;
        __syncthreads();
        for (int s = 128; s > 0; s >>= 1) {
            if (tid < s) {
                if (redv[tid + s] > redv[tid] ||
                    (redv[tid + s] == redv[tid] && redi[tid + s] < redi[tid])) {
                    redv[tid] = redv[tid + s]; redi[tid] = redi[tid + s];
                }
            }
            __syncthreads();
        }
        if (tid == 0) s_last = redi[0];
        __syncthreads();
    }
}

__global__ void gather_node(const float* __restrict__ x, const int* __restrict__ fidx,
                            float* __restrict__ out)
{
    int t = blockIdx.x * 256 + threadIdx.x;
    if (t >= B_ * 3 * M_) return;
    int b = t / (3 * M_), r = t % (3 * M_), c = r / M_, m = r % M_;
    out[t] = x[(long long)b * 3 * N_ + (long long)c * N_ + fidx[b * M_ + m]];
}

__global__ void gather_nf1(const float* __restrict__ x12, const int* __restrict__ fidx,
                           float* __restrict__ xm)
{
    long long t = (long long)blockIdx.x * 256 + threadIdx.x;
    if (t >= (long long)B_ * 128 * M_) return;
    int b = (int)(t / (128 * M_)); int r = (int)(t % (128 * M_));
    int c = r / M_, m = r % M_;
    xm[(long long)b * 256 * M_ + (long long)c * M_ + m] =
        x12[(long long)b * 128 * N_ + (long long)c * N_ + fidx[b * M_ + m]];
}

__global__ void aggmax_kernel(const float* __restrict__ x12, const int* __restrict__ idx,
                              float* __restrict__ xm)
{
    long long t = (long long)blockIdx.x * 256 + threadIdx.x;
    if (t >= (long long)B_ * 128 * M_) return;
    int b = (int)(t / (128 * M_)); int r = (int)(t % (128 * M_));
    int c = r / M_, m = r % M_;
    const int* ib = idx + (long long)b * M_ * K_ + (long long)m * K_;
    float mx = NEG_INF_;
    for (int j = 0; j < K_; ++j)
        mx = fmaxf(mx, x12[(long long)b * 128 * N_ + (long long)c * N_ + ib[j]]);
    xm[(long long)b * 256 * M_ + (long long)(128 + c) * M_ + m] = mx;
}

__global__ void init_v(float* __restrict__ v)
{
    int t = blockIdx.x * 256 + threadIdx.x;
    if (t < B_ * 2 * E_) v[t] = NEG_INF_;
}

// ---------------------------------------------------------------------------
extern "C" void kernel_launch(void* const* d_in, const int* in_sizes, int n_in,
                              void* d_out, int out_size, void* d_ws, size_t ws_size,
                              hipStream_t stream)
{
    (void)in_sizes; (void)n_in; (void)out_size; (void)ws_size;
    const float* x   = (const float*)d_in[0];
    const float* W1  = (const float*)d_in[1];
    const float* g1  = (const float*)d_in[2];
    const float* b1  = (const float*)d_in[3];
    const float* W2  = (const float*)d_in[4];
    const float* g2  = (const float*)d_in[5];
    const float* b2  = (const float*)d_in[6];
    const float* W2m = (const float*)d_in[7];
    const float* g2m = (const float*)d_in[8];
    const float* b2m = (const float*)d_in[9];
    const float* W3  = (const float*)d_in[10];
    const float* g3  = (const float*)d_in[11];
    const float* b3  = (const float*)d_in[12];
    const float* W4  = (const float*)d_in[13];
    const float* g4  = (const float*)d_in[14];
    const float* b4  = (const float*)d_in[15];
    const float* W5  = (const float*)d_in[16];
    const float* g5  = (const float*)d_in[17];
    const float* b5  = (const float*)d_in[18];
    const float* Wl1 = (const float*)d_in[19];
    const float* g6  = (const float*)d_in[20];
    const float* b6  = (const float*)d_in[21];
    const float* Wl2 = (const float*)d_in[22];
    const float* bl2 = (const float*)d_in[23];
    const float* g7  = (const float*)d_in[24];
    const float* b7  = (const float*)d_in[25];
    const float* Wl3 = (const float*)d_in[26];
    const float* bl3 = (const float*)d_in[27];
    float* out = (float*)d_out;

    // ---- workspace carve ----
    char* p = (char*)d_ws;
    auto alloc = [&](size_t bytes) {
        void* r = (void*)p;
        p += (bytes + 255) & ~(size_t)255;
        return r;
    };
    float* tu   = (float*)alloc((size_t)B_ * 128 * N_ * 4);  // reused (B*512*512 == B*128*2048)
    float* x12  = (float*)alloc((size_t)B_ * 128 * N_ * 4);
    float* xm   = (float*)alloc((size_t)B_ * 256 * M_ * 4);
    float* x34  = (float*)alloc((size_t)B_ * 512 * M_ * 4);
    float* Abuf = (float*)alloc((size_t)512 * 256 * 4);
    int*   idxb = (int*)alloc((size_t)B_ * N_ * K_ * 4);
    int*   fidx = (int*)alloc((size_t)B_ * M_ * 4);
    float* v    = (float*)alloc((size_t)B_ * 2 * E_ * 4);
    float* h1   = (float*)alloc((size_t)512 * B_ * 4);
    float* h2   = (float*)alloc((size_t)256 * B_ * 4);

    auto cdiv = [](long long a, long long b) { return (int)((a + b - 1) / b); };

    init_v<<<cdiv(B_ * 2 * E_, 256), 256, 0, stream>>>(v);

    // ---- edge_conv 1 (x: C=3, N=2048, k=20) ----
    knn_kernel<<<dim3(cdiv(N_, 128), B_), 128, 0, stream>>>(
        x, 3LL * N_, x, 3LL * N_, 3, N_, N_, K_, idxb, (long long)N_ * K_);
    prep_edge_w<<<cdiv(128 * 3, 256), 256, 0, stream>>>(W1, Abuf, 64, 3);
    gemm16_wmma<<<dim3(N_ / 32, 2, B_), 32, 0, stream>>>(
        Abuf, 128, 3, x, 3LL * N_, N_, 1, N_,
        tu, 128LL * N_, N_, 1, nullptr, nullptr, nullptr, nullptr, 0, 0);
    edge_gathermax<<<dim3(cdiv(64LL * N_, 256), B_), 256, 0, stream>>>(
        tu, 128LL * N_, 64, N_, idxb, (long long)N_ * K_, K_, g1, b1, x12, 128LL * N_, N_);

    // ---- edge_conv 2 (x1: C=64) -> x12 rows 64..127 ----
    knn_kernel<<<dim3(cdiv(N_, 128), B_), 128, 0, stream>>>(
        x12, 128LL * N_, x12, 128LL * N_, 64, N_, N_, K_, idxb, (long long)N_ * K_);
    prep_edge_w<<<cdiv(128 * 64, 256), 256, 0, stream>>>(W2, Abuf, 64, 64);
    gemm16_wmma<<<dim3(N_ / 32, 2, B_), 32, 0, stream>>>(
        Abuf, 128, 64, x12, 128LL * N_, N_, 1, N_,
        tu, 128LL * N_, N_, 1, nullptr, nullptr, nullptr, nullptr, 0, 0);
    edge_gathermax<<<dim3(cdiv(64LL * N_, 256), B_), 256, 0, stream>>>(
        tu, 128LL * N_, 64, N_, idxb, (long long)N_ * K_, K_, g2, b2,
        x12 + 64LL * N_, 128LL * N_, N_);

    // ---- xt1 = lrelu(bn(W2m @ x12)); v1 = max over n (fused atomic-max) ----
    gemm16_wmma<<<dim3(N_ / 32, E_ / 64, B_), 32, 0, stream>>>(
        W2m, E_, 128, x12, 128LL * N_, N_, 1, N_,
        tu, 0, 0, 0, g2m, b2m, nullptr, v, 2LL * E_, 2);

    // ---- FPS + gathers + aggregate ----
    fps_kernel<<<B_, 256, 0, stream>>>(x, fidx);
    gather_node<<<cdiv(B_ * 3 * M_, 256), 256, 0, stream>>>(x, fidx, out + B_ * 40);
    gather_nf1<<<cdiv((long long)B_ * 128 * M_, 256), 256, 0, stream>>>(x12, fidx, xm);
    knn_kernel<<<dim3(cdiv(M_, 128), B_), 128, 0, stream>>>(
        x, 3LL * N_, out + B_ * 40, 3LL * M_, 3, N_, M_, K_, idxb, (long long)M_ * K_);
    aggmax_kernel<<<cdiv((long long)B_ * 128 * M_, 256), 256, 0, stream>>>(x12, idxb, xm);

    // ---- edge_conv 3 (xm: C=256, N=512, k=10) -> x34 rows 0..255 ----
    knn_kernel<<<dim3(cdiv(M_, 128), B_), 128, 0, stream>>>(
        xm, 256LL * M_, xm, 256LL * M_, 256, M_, M_, K2_, idxb, (long long)M_ * K2_);
    prep_edge_w<<<cdiv(512 * 256, 256), 256, 0, stream>>>(W3, Abuf, 256, 256);
    gemm16_wmma<<<dim3(M_ / 32, 8, B_), 32, 0, stream>>>(
        Abuf, 512, 256, xm, 256LL * M_, M_, 1, M_,
        tu, 512LL * M_, M_, 1, nullptr, nullptr, nullptr, nullptr, 0, 0);
    edge_gathermax<<<dim3(cdiv(256LL * M_, 256), B_), 256, 0, stream>>>(
        tu, 512LL * M_, 256, M_, idxb, (long long)M_ * K2_, K2_, g3, b3, x34, 512LL * M_, M_);

    // ---- edge_conv 4 (x3: C=256) -> x34 rows 256..511 ----
    knn_kernel<<<dim3(cdiv(M_, 128), B_), 128, 0, stream>>>(
        x34, 512LL * M_, x34, 512LL * M_, 256, M_, M_, K2_, idxb, (long long)M_ * K2_);
    prep_edge_w<<<cdiv(512 * 256, 256), 256, 0, stream>>>(W4, Abuf, 256, 256);
    gemm16_wmma<<<dim3(M_ / 32, 8, B_), 32, 0, stream>>>(
        Abuf, 512, 256, x34, 512LL * M_, M_, 1, M_,
        tu, 512LL * M_, M_, 1, nullptr, nullptr, nullptr, nullptr, 0, 0);
    edge_gathermax<<<dim3(cdiv(256LL * M_, 256), B_), 256, 0, stream>>>(
        tu, 512LL * M_, 256, M_, idxb, (long long)M_ * K2_, K2_, g4, b4,
        x34 + 256LL * M_, 512LL * M_, M_);

    // ---- xt2 = lrelu(bn(W5 @ x34)); v2 = max over n ----
    gemm16_wmma<<<dim3(M_ / 32, E_ / 64, B_), 32, 0, stream>>>(
        W5, E_, 512, x34, 512LL * M_, M_, 1, M_,
        tu, 0, 0, 0, g5, b5, nullptr, v + E_, 2LL * E_, 2);

    // ---- FC head (cols = batch = 8, padded in 16-wide tiles) ----
    gemm16_wmma<<<dim3(1, 8, 1), 32, 0, stream>>>(
        Wl1, 512, 2 * E_, v, 0, 1, 2LL * E_, B_,
        h1, 0, B_, 1, g6, b6, nullptr, nullptr, 0, 1);
    gemm16_wmma<<<dim3(1, 4, 1), 32, 0, stream>>>(
        Wl2, 256, 512, h1, 0, B_, 1, B_,
        h2, 0, B_, 1, g7, b7, bl2, nullptr, 0, 1);
    gemm16_wmma<<<dim3(1, 1, 1), 32, 0, stream>>>(
        Wl3, 40, 256, h2, 0, B_, 1, B_,
        out, 0, 1, 40, nullptr, nullptr, bl3, nullptr, 0, 3);
}